// GCN_8907762172440
// MI455X (gfx1250) — compile-verified
//
#include <hip/hip_runtime.h>
#include <hip/hip_bf16.h>

typedef __attribute__((ext_vector_type(2))) float v2f;
typedef __attribute__((ext_vector_type(8))) float v8f;

#define NNODES 50000
#define NEDGES 800000
#define NGRAPH 512
#define INDIM  128
#define HID    96
#define OUTDIM 10
#define NCONV  4
#define NTILE  6      // 96 / 16 output column tiles
#define GEMM_WAVES 8  // waves (m-tiles) per block

static inline int cdiv(int a, int b) { return (a + b - 1) / b; }

// ---------------- degree / norm ----------------
__global__ void k_deg_init(float* deg, int n) {
    int i = blockIdx.x * blockDim.x + threadIdx.x;
    if (i < n) deg[i] = 1.0f;                 // self-loop contributes 1
}

__global__ void k_deg_accum(const int* __restrict__ ei, float* deg, int E) {
    int e = blockIdx.x * blockDim.x + threadIdx.x;
    if (e < E) atomicAdd(&deg[ei[E + e]], 1.0f);   // dst row of edge_index
}

__global__ void k_rsqrt_inplace(float* deg, int n) {
    int i = blockIdx.x * blockDim.x + threadIdx.x;
    if (i < n) deg[i] = rsqrtf(deg[i]);
}

// ---------------- WMMA fp32 GEMM ----------------
// C[M,96] = (RELUA ? relu(A) : A)[M,K] @ B[K,96]
// One wave computes a 16x96 strip (6 tiles, A fragment reused 6x per k-step).
// B staged in LDS transposed with +2 padding: sBt[col*(K+2)+k], so each B
// fragment is one aligned ds_load_b64 and half-wave lanes hit distinct banks.
// All mode flags are template params so no runtime selects pollute the loop.
// Epilogues:
//   FUSE=0: C = acc + bias (RELU optional)
//   FUSE=1: C = acc (raw hw), C2 = acc*dinv[row]^2 + bias  (GCN self-loop init)
template <int RELUA, int RELU, int FUSE>
__global__ void k_gemm_wmma(const float* __restrict__ A, const float* __restrict__ B,
                            const float* __restrict__ bias,
                            float* __restrict__ C, float* __restrict__ C2,
                            const float* __restrict__ dinv,
                            int M, int K, int lda) {
    extern __shared__ float sBt[];             // [96, K+2] transposed, padded
    int lstride = K + 2;

    // cooperative transpose-stage of B into LDS (all waves, then sync)
    int nB = K * HID;
    for (int i = threadIdx.x; i < nB; i += blockDim.x) {
        int k   = i / HID;
        int col = i - k * HID;
        sBt[col * lstride + k] = B[i];
    }
    __syncthreads();

    int mt   = blockIdx.x * GEMM_WAVES + (threadIdx.x >> 5);
    int lane = threadIdx.x & 31;
    int mTiles = M >> 4;
    if (mt >= mTiles) return;                  // wave-uniform exit after barrier

    int half = lane >> 4;                      // 0: K pair {k,k+1}, 1: {k+2,k+3}
    int l15  = lane & 15;

    const float* arow  = A + (size_t)(mt * 16 + l15) * lda;  // A: lane&15 = M row
    const float* bbase = sBt + (size_t)l15 * lstride;

    v8f acc[NTILE] = {};
    for (int k = 0; k < K; k += 4) {
        int kb = k + half * 2;
        v2f a = *(const v2f*)(arow + kb);                    // 8B-aligned (lda even)
        if (RELUA) { a[0] = fmaxf(a[0], 0.0f); a[1] = fmaxf(a[1], 0.0f); }
        #pragma unroll
        for (int nt = 0; nt < NTILE; nt++) {
            v2f b = *(const v2f*)(bbase + (size_t)(nt * 16) * lstride + kb);
            acc[nt] = __builtin_amdgcn_wmma_f32_16x16x4_f32(
                /*neg_a=*/false, a, /*neg_b=*/false, b,
                /*c_mod=*/(short)0, acc[nt], /*reuse_a=*/false, /*reuse_b=*/false);
        }
    }

    int rbase = mt * 16 + half * 8;            // first output row for this lane
    float* cbase = C + (size_t)rbase * HID + l15;

    if (FUSE) {                                // fused GCN self-loop init
        float dv2[8];
        #pragma unroll
        for (int v = 0; v < 8; v++) { float d = dinv[rbase + v]; dv2[v] = d * d; }
        float* c2base = C2 + (size_t)rbase * HID + l15;
        #pragma unroll
        for (int nt = 0; nt < NTILE; nt++) {
            float bv = bias[nt * 16 + l15];
            #pragma unroll
            for (int v = 0; v < 8; v++) {
                float raw = acc[nt][v];
                cbase [(size_t)v * HID + nt * 16] = raw;                 // hw
                c2base[(size_t)v * HID + nt * 16] = raw * dv2[v] + bv;   // self + b
            }
        }
    } else {
        #pragma unroll
        for (int nt = 0; nt < NTILE; nt++) {
            float bv = bias[nt * 16 + l15];
            #pragma unroll
            for (int v = 0; v < 8; v++) {
                float r = acc[nt][v] + bv;
                if (RELU) r = fmaxf(r, 0.0f);
                cbase[(size_t)v * HID + nt * 16] = r;
            }
        }
    }
}

// ---------------- GCN aggregation ----------------
// wave-per-edge: out[dst, :] += hw[src, :] * dinv[src]*dinv[dst]  (96 = 3 x 32)
__global__ void k_edge_scatter(const float* __restrict__ hw, const float* __restrict__ dinv,
                               const int* __restrict__ ei, float* __restrict__ out, int E) {
    int wave = (blockIdx.x * blockDim.x + threadIdx.x) >> 5;
    int lane = threadIdx.x & 31;
    if (wave >= E) return;
    int s = ei[wave];
    int d = ei[E + wave];
    float norm = dinv[s] * dinv[d];
    const float* srow = hw  + (size_t)s * HID;
    float*       drow = out + (size_t)d * HID;
    #pragma unroll
    for (int k = 0; k < 3; k++) {
        int c = lane + k * 32;
        atomicAdd(drow + c, srow[c] * norm);
    }
}

// ---------------- pooling + decoder tail ----------------
__global__ void k_zero(float* x, int n) {
    int i = blockIdx.x * blockDim.x + threadIdx.x;
    if (i < n) x[i] = 0.0f;
}

// g[batch[i], c] += relu(h[i, c])   (relu fused on read: h is pre-relu)
__global__ void k_pool(const float* __restrict__ h, const int* __restrict__ batch,
                       float* __restrict__ g, int n) {
    int idx = blockIdx.x * blockDim.x + threadIdx.x;
    if (idx < n * HID) {
        int node = idx / HID;
        int c    = idx - node * HID;
        atomicAdd(&g[(size_t)batch[node] * HID + c], fmaxf(h[idx], 0.0f));
    }
}

// out[g, o] = g2[g, :] . W1[:, o] + b1[o]   (512 x 10)
__global__ void k_dec2(const float* __restrict__ g2, const float* __restrict__ W,
                       const float* __restrict__ b, float* __restrict__ out) {
    int idx = blockIdx.x * blockDim.x + threadIdx.x;
    if (idx < NGRAPH * OUTDIM) {
        int gi = idx / OUTDIM;
        int o  = idx - gi * OUTDIM;
        float s = b[o];
        const float* row = g2 + (size_t)gi * HID;
        #pragma unroll 8
        for (int k = 0; k < HID; k++) s += row[k] * W[k * OUTDIM + o];
        out[idx] = s;
    }
}

extern "C" void kernel_launch(void* const* d_in, const int* in_sizes, int n_in,
                              void* d_out, int out_size, void* d_ws, size_t ws_size,
                              hipStream_t stream) {
    const float* x       = (const float*)d_in[0];
    const int*   ei      = (const int*)  d_in[1];   // [2, E] int32
    const int*   batch   = (const int*)  d_in[2];   // [N] int32
    const float* enc_W0  = (const float*)d_in[3];
    const float* enc_b0  = (const float*)d_in[4];
    const float* enc_W1  = (const float*)d_in[5];
    const float* enc_b1  = (const float*)d_in[6];
    const float* conv_W  = (const float*)d_in[7];   // [4, 96, 96]
    const float* conv_b  = (const float*)d_in[8];   // [4, 96]
    const float* dec_W0  = (const float*)d_in[9];
    const float* dec_b0  = (const float*)d_in[10];
    const float* dec_W1  = (const float*)d_in[11];
    const float* dec_b1  = (const float*)d_in[12];
    float* out = (float*)d_out;

    // workspace layout (floats)
    float* ws   = (float*)d_ws;
    size_t feat = (size_t)NNODES * HID;             // 4.8M floats each
    float* buf0 = ws;                               // hw / encoder temp
    float* buf1 = buf0 + feat;                      // h ping
    float* buf2 = buf1 + feat;                      // h pong
    float* dinv = buf2 + feat;                      // [N]
    float* g    = dinv + NNODES;                    // [512, 96]
    float* g2   = g + (size_t)NGRAPH * HID;         // [512, 96]

    const int B = 256;

    // 1. deg / dinv
    k_deg_init<<<cdiv(NNODES, B), B, 0, stream>>>(dinv, NNODES);
    k_deg_accum<<<cdiv(NEDGES, B), B, 0, stream>>>(ei, dinv, NEDGES);
    k_rsqrt_inplace<<<cdiv(NNODES, B), B, 0, stream>>>(dinv, NNODES);

    // 2. encoder MLP
    int mTilesN   = NNODES / 16;                    // 3125
    int gemmBlksN = cdiv(mTilesN, GEMM_WAVES);
    size_t shm128 = (size_t)HID * (INDIM + 2) * sizeof(float);  // ~49.9 KB
    size_t shm96  = (size_t)HID * (HID   + 2) * sizeof(float);  // ~36.8 KB
    k_gemm_wmma<0, 1, 0><<<gemmBlksN, B, shm128, stream>>>(
        x, enc_W0, enc_b0, buf0, nullptr, nullptr, NNODES, INDIM, INDIM);
    k_gemm_wmma<0, 0, 0><<<gemmBlksN, B, shm96, stream>>>(
        buf0, enc_W1, enc_b1, buf1, nullptr, nullptr, NNODES, HID, HID);

    // 3. GCN conv layers: GEMM (fused self-loop init) + edge scatter.
    //    relu applied lazily on read (RELUA for l>0; pool for the last layer).
    float* cur = buf1;
    float* nxt = buf2;
    int edgeBlks = cdiv(NEDGES * 32, B);            // wave per edge
    for (int l = 0; l < NCONV; l++) {
        const float* Wl = conv_W + (size_t)l * HID * HID;
        const float* bl = conv_b + (size_t)l * HID;
        if (l == 0)
            k_gemm_wmma<0, 0, 1><<<gemmBlksN, B, shm96, stream>>>(
                cur, Wl, bl, buf0 /*hw*/, nxt /*self-init*/, dinv, NNODES, HID, HID);
        else
            k_gemm_wmma<1, 0, 1><<<gemmBlksN, B, shm96, stream>>>(
                cur, Wl, bl, buf0 /*hw*/, nxt /*self-init*/, dinv, NNODES, HID, HID);
        k_edge_scatter<<<edgeBlks, B, 0, stream>>>(buf0, dinv, ei, nxt, NEDGES);
        float* t = cur; cur = nxt; nxt = t;
    }

    // 4. global add pool (relu fused on read)
    k_zero<<<cdiv(NGRAPH * HID, B), B, 0, stream>>>(g, NGRAPH * HID);
    k_pool<<<cdiv(NNODES * HID, B), B, 0, stream>>>(cur, batch, g, NNODES);

    // 5. decoder MLP
    int mTilesG = NGRAPH / 16;                      // 32
    k_gemm_wmma<0, 1, 0><<<cdiv(mTilesG, GEMM_WAVES), B, shm96, stream>>>(
        g, dec_W0, dec_b0, g2, nullptr, nullptr, NGRAPH, HID, HID);
    k_dec2<<<cdiv(NGRAPH * OUTDIM, B), B, 0, stream>>>(g2, dec_W1, dec_b1, out);
}